// CritiGraph_64175401337324
// MI455X (gfx1250) — compile-verified
//
#include <hip/hip_runtime.h>

// Problem constants from the reference
#define HBITS 16
#define NK    2
#define TPk   8
#define CURTP 4
#define Sk    64
#define Dk    256
#define Ck    65           // 2*H*N_K + 1
#define TSTRIDE (Dk + 4)   // padded LDS row pitch (floats): bank-conflict-free WMMA reads

typedef float v2f __attribute__((ext_vector_type(2)));
typedef float v4f __attribute__((ext_vector_type(4)));
typedef float v8f __attribute__((ext_vector_type(8)));

#if defined(__gfx1250__) && __has_builtin(__builtin_amdgcn_tensor_load_to_lds)
#define HAVE_TDM 1
typedef unsigned int u32x4 __attribute__((ext_vector_type(4)));
typedef int          i32x4 __attribute__((ext_vector_type(4)));
typedef int          i32x8 __attribute__((ext_vector_type(8)));
#endif

// sign(a)*sign(b) * (1 - bitlen(|a|^|b|+1)/16); exact vs the reference's
// frexp form since (|a|^|b|)+1 <= 2^18 is exactly representable in f32.
__device__ __forceinline__ float cg_cos_sim(int a, int b) {
    float sg = ((a >= 0) == (b >= 0)) ? 1.0f : -1.0f;
    int aa = a < 0 ? -a : a;
    int bb = b < 0 ? -b : b;
    int x = (aa ^ bb) + 1;
    int e = 32 - __clz(x);
    return sg * (1.0f - (float)e * 0.0625f);
}

__global__ __launch_bounds__(256) void critigraph_kernel(
    const int*   __restrict__ sta_loc,      // (T, TP)
    const int*   __restrict__ nei_loc,      // (T, S, TP)
    const int*   __restrict__ rand_numbers, // (T, H, NK, TP)
    const float* __restrict__ sta_emb,      // (T, D)
    const float* __restrict__ nei_emb,      // (T, S, D)
    const float* __restrict__ mask,         // (T, S)
    const float* __restrict__ rand_vals,    // (T, TP)
    const float* __restrict__ t_rand,       // (T,)
    float* __restrict__ out_loc,            // (T, TP) selected_locs (as f32)
    float* __restrict__ out_loss)           // (T,)    real_loss
{
    const int t    = blockIdx.x;
    const int tid  = threadIdx.x;
    const int lane = tid & 31;
    const int wave = tid >> 5;

    __shared__ float s_tile[Sk * TSTRIDE];   // 64x256 nei_emb tile, padded rows
    __shared__ float s_sta[Dk];
    __shared__ float s_dot[Sk];
    __shared__ float s_norm2[Sk];
    __shared__ float s_stanorm2;
    __shared__ float s_lth;
    __shared__ int   s_staloc[TPk];
    __shared__ int   s_nei[Sk * TPk];
    __shared__ int   s_cnc[Ck * TPk];
    __shared__ float s_u[Sk * TPk];
    __shared__ float s_w[Sk];
    __shared__ float s_mask[Sk];
    __shared__ float s_loss[Ck * TPk];
    __shared__ float s_rv[TPk];
    __shared__ int   s_amin[TPk];
    __shared__ float s_selloss[TPk];

    const int swave = __builtin_amdgcn_readfirstlane(wave);  // scalar wave id

#ifdef HAVE_TDM
    // ---- TDM: one tensor_load_to_lds streams the 64KB nei_emb tile -------
    // Scalar branch: only wave 0 reaches the instruction (TDM ignores EXEC).
    if (swave == 0) {
        unsigned long long ga =
            (unsigned long long)(uintptr_t)(nei_emb + (size_t)t * Sk * Dk);
        unsigned ldsa = (unsigned)(uintptr_t)(&s_tile[0]);  // low 32b = LDS offset
        u32x4 g0;
        g0[0] = 1u;                                    // count=1 (valid), no gather
        g0[1] = ldsa;                                  // lds_addr
        g0[2] = (unsigned)ga;                          // global_addr[31:0]
        g0[3] = (unsigned)((ga >> 32) & 0x01FFFFFFu)   // global_addr[56:32]
              | (2u << 30);                            // type=2 (image)
        i32x8 g1;
        // data_size=4B(2)<<16 | pad_enable<<20 | pad_interval=7(256 dw)<<22
        // | pad_amount=3(4 dw)<<25  -> +16B per 1KB row => TSTRIDE pitch
        g1[0] = (int)((2u << 16) | (1u << 20) | (7u << 22) | (3u << 25));
        g1[1] = (int)(256u << 16);   // tensor_dim0[15:0]=256 in bits[31:16]
        g1[2] = (int)(64u << 16);    // tensor_dim0[31:16]=0, tensor_dim1=64
        g1[3] = (int)(256u << 16);   // tensor_dim1 hi=0, tile_dim0=256
        g1[4] = 64;                  // tile_dim1=64, tile_dim2=0
        g1[5] = 256;                 // tensor_dim0_stride[31:0]=256
        g1[6] = 0;                   // stride hi, dim1_stride lo
        g1[7] = 0;
        i32x4 gz  = {0, 0, 0, 0};
        i32x8 gz8 = {0, 0, 0, 0, 0, 0, 0, 0};
        // 6-arg form (clang-23 / therock-10.0 headers)
        __builtin_amdgcn_tensor_load_to_lds(g0, g1, gz, gz, gz8, 0);
    }
#endif

    // ---- phase 0a: zero accumulators -------------------------------------
    if (tid < Sk) { s_norm2[tid] = 0.0f; s_dot[tid] = 0.0f; }
    if (tid == 0) { s_stanorm2 = 0.0f; s_lth = 1e-12f; }
    __syncthreads();

    // ---- phase 0b: stage small inputs, partial reductions ----------------
    {
        float sv = sta_emb[(size_t)t * Dk + tid];  // 256 threads == D
        s_sta[tid] = sv;
        atomicAdd(&s_stanorm2, sv * sv);           // ds_add_f32
    }
    for (int i = tid; i < Sk * TPk; i += 256)
        s_nei[i] = nei_loc[(size_t)t * Sk * TPk + i];
    if (tid < Sk) {
        float m = mask[t * Sk + tid];
        s_mask[tid] = m;
        atomicAdd(&s_lth, m);
    }
    if (tid < TPk) s_staloc[tid] = sta_loc[t * TPk + tid];

#ifndef HAVE_TDM
    // Fallback staging: coalesced b128 copy with padded row pitch
    {
        const v4f* __restrict__ gp = (const v4f*)(nei_emb + (size_t)t * Sk * Dk);
        for (int i = tid; i < (Sk * Dk) / 4; i += 256) {
            int row = i >> 6;              // /(Dk/4)
            int col = (i & 63) << 2;
            *(v4f*)(&s_tile[row * TSTRIDE + col]) = gp[i];
        }
    }
#else
    if (swave == 0) __builtin_amdgcn_s_wait_tensorcnt(0);  // tile in LDS
#endif
    __syncthreads();

    // ---- phase 0c: candidate locations cnc_loc (65 x TP) -----------------
    {
        int i = tid;                 // exactly 256 = 32*TP entries
        int c = i >> 3, p = i & 7;
        int h = c / NK, j = c % NK;
        int r = rand_numbers[(((size_t)t * HBITS + h) * NK + j) * TPk + p];
        int v = (s_staloc[p] ^ (1 << h)) ^ (r & ((1 << h) - 1));
        s_cnc[c * TPk + p]        = v;    // res
        s_cnc[(33 + c) * TPk + p] = -v;   // -res
        if (tid < TPk) s_cnc[32 * TPk + tid] = s_staloc[tid];  // ori
    }

    // ---- phase 1: fused WMMA dots + neighbor norms, all from LDS ---------
    // 8 waves = 4 neighbor-blocks x 2 K-halves.
    // A (16x4 f32): sta chunk broadcast into all rows  -> ds_load_b64 (s_sta)
    // B (4x16 f32): column n = neighbor sblk+n         -> ds_load_b64 (s_tile)
    // Padded pitch => lane banks 4*(l%16)(+2 for upper half): conflict-free.
    {
        const int sblk  = (wave & 3) * 16;
        const int kbase = (wave >> 2) * (Dk / 2);
        const int srow  = sblk + (lane & 15);
        const int coff  = (lane >> 4) * 2;  // 0 or 2: K-pair within chunk
        const float* brow = s_tile + srow * TSTRIDE + coff;
        const float* arow = s_sta + coff;
        v8f  acc = {};
        float np = 0.0f;
#pragma unroll 4
        for (int k = kbase; k < kbase + Dk / 2; k += 4) {
            v2f a = *(const v2f*)(arow + k);
            v2f b = *(const v2f*)(brow + k);
            np  = fmaf(b[0], b[0], np);
            np  = fmaf(b[1], b[1], np);
            acc = __builtin_amdgcn_wmma_f32_16x16x4_f32(
                      false, a, false, b, (short)0, acc, false, false);
        }
        atomicAdd(&s_norm2[srow], np);                  // both K-pairs + halves
        if (lane < 16) atomicAdd(&s_dot[srow], acc[0]); // rows duplicate: one add
    }
    __syncthreads();

    // ---- phase 2: eu, weights w, cos_sn, u (one thread per s) ------------
    if (tid < Sk) {
        float cs[TPk];
        float sum = 0.0f;
#pragma unroll
        for (int p = 0; p < TPk; ++p) {
            cs[p] = cg_cos_sim(s_staloc[p], s_nei[tid * TPk + p]);
            sum += cs[p];
        }
        float eu = s_dot[tid] / sqrtf(s_norm2[tid] * s_stanorm2);
        s_w[tid] = fabsf(eu) * s_mask[tid] / s_lth;
#pragma unroll
        for (int p = 0; p < TPk; ++p)
            s_u[tid * TPk + p] = (sum - cs[p]) * 0.125f - eu;
    }
    __syncthreads();

    // ---- phase 3: loss[c][p] = sum_s w[s] * (u[s][p] + cos_cn/8)^2 -------
    for (int idx = tid; idx < Ck * TPk; idx += 256) {
        int p = idx & 7;
        int a = s_cnc[idx];
        float acc = 0.0f;
        for (int s = 0; s < Sk; ++s) {
            float cc = cg_cos_sim(a, s_nei[s * TPk + p]);
            float d  = fmaf(cc, 0.125f, s_u[s * TPk + p]);
            acc = fmaf(s_w[s] * d, d, acc);
        }
        s_loss[idx] = acc;
    }
    __syncthreads();

    // ---- phase 4: argmin over candidates (first-min, like jnp.argmin) ----
    if (tid < TPk) {
        float best = s_loss[tid];
        int   bc   = 0;
        for (int c = 1; c < Ck; ++c) {
            float v = s_loss[c * TPk + tid];
            if (v < best) { best = v; bc = c; }
        }
        s_amin[tid] = bc;
        s_rv[tid]   = rand_vals[t * TPk + tid];
    }
    __syncthreads();

    // ---- phase 5: stable top-CUR_TP selection + outputs ------------------
    if (tid < TPk) {
        float rv   = s_rv[tid];
        int   rank = 0;
#pragma unroll
        for (int p = 0; p < TPk; ++p) {
            float o = s_rv[p];
            rank += (o < rv || (o == rv && p < tid)) ? 1 : 0;  // stable argsort
        }
        bool chosen = (rank < CURTP) && (t_rand[t] < 0.8f);
        int  ci     = chosen ? s_amin[tid] : (HBITS * NK);     // 32 == ori
        out_loc[t * TPk + tid] = (float)s_cnc[ci * TPk + tid];
        s_selloss[tid]         = s_loss[ci * TPk + tid];
    }
    __syncthreads();
    if (tid == 0) {
        float rl = 0.0f;
#pragma unroll
        for (int p = 0; p < TPk; ++p) rl += s_selloss[p];
        out_loss[t] = rl * (1.0f / TPk);
    }
}

extern "C" void kernel_launch(void* const* d_in, const int* in_sizes, int n_in,
                              void* d_out, int out_size, void* d_ws, size_t ws_size,
                              hipStream_t stream) {
    const int*   sta_loc      = (const int*)d_in[0];
    const int*   nei_loc      = (const int*)d_in[1];
    const int*   rand_numbers = (const int*)d_in[2];
    const float* sta_emb      = (const float*)d_in[3];
    const float* nei_emb      = (const float*)d_in[4];
    const float* mask         = (const float*)d_in[5];
    const float* rand_vals    = (const float*)d_in[6];
    const float* t_rand       = (const float*)d_in[7];

    const int T = in_sizes[7];          // t_rand length
    float* out      = (float*)d_out;    // [T*TP] selected_locs, then [T] loss
    float* out_loss = out + (size_t)T * TPk;

    critigraph_kernel<<<T, 256, 0, stream>>>(
        sta_loc, nei_loc, rand_numbers, sta_emb, nei_emb, mask,
        rand_vals, t_rand, out, out_loss);
}